// QGCN_38843684225695
// MI455X (gfx1250) — compile-verified
//
#include <hip/hip_runtime.h>
#include <hip/hip_bf16.h>

typedef float v2f __attribute__((ext_vector_type(2)));
typedef float v8f __attribute__((ext_vector_type(8)));

// ---------------------------------------------------------------------------
// GEMM  C[M,N] = A[M,K] * W[K,N]   (f32, row-major), via V_WMMA_F32_16X16X4_F32
// One wave (32 lanes) computes a 16x64 output strip: 4 16x16 accumulators,
// A-fragment reused across the 4 WMMAs of each k-step.
//
// Fragment layouts per CDNA5 ISA 7.12.2 (wave32):
//   A (16x4):  lane l: row = l&15; VGPR0 = K(2*(l>>4)), VGPR1 = K(2*(l>>4)+1)
//   B (4x16):  lane l: col = l&15; VGPR0 = K(2*(l>>4)), VGPR1 = K(2*(l>>4)+1)
//   C/D (16x16): lane l: col = l&15; VGPR r: row = (l>>4)*8 + r
// ---------------------------------------------------------------------------
__global__ __launch_bounds__(32)
void gemm_wmma_f32(const float* __restrict__ A, const float* __restrict__ W,
                   float* __restrict__ C, int M, int K, int N)
{
    const int tilesN = N >> 6;                 // groups of 64 columns
    const int tm = blockIdx.x / tilesN;        // 16-row tile index
    const int tg = blockIdx.x % tilesN;        // 64-col group index
    const int lane = threadIdx.x;
    const int half = lane >> 4;                // 0 or 1
    const int l15  = lane & 15;

    const float* Arow  = A + (size_t)(tm * 16 + l15) * K + 2 * half;
    const float* Bbase = W + tg * 64 + l15;

    v8f acc0 = {}, acc1 = {}, acc2 = {}, acc3 = {};

    for (int k0 = 0; k0 < K; k0 += 4) {
        v2f a;
        a.x = Arow[k0];
        a.y = Arow[k0 + 1];

        const float* Brow0 = Bbase + (size_t)(k0 + 2 * half) * N;
        const float* Brow1 = Brow0 + N;

        v2f b0, b1, b2, b3;
        b0.x = Brow0[0];   b0.y = Brow1[0];
        b1.x = Brow0[16];  b1.y = Brow1[16];
        b2.x = Brow0[32];  b2.y = Brow1[32];
        b3.x = Brow0[48];  b3.y = Brow1[48];

        acc0 = __builtin_amdgcn_wmma_f32_16x16x4_f32(false, a, false, b0, (short)0, acc0, false, false);
        acc1 = __builtin_amdgcn_wmma_f32_16x16x4_f32(false, a, false, b1, (short)0, acc1, false, false);
        acc2 = __builtin_amdgcn_wmma_f32_16x16x4_f32(false, a, false, b2, (short)0, acc2, false, false);
        acc3 = __builtin_amdgcn_wmma_f32_16x16x4_f32(false, a, false, b3, (short)0, acc3, false, false);
    }

    float* Crow = C + (size_t)(tm * 16 + half * 8) * N + tg * 64 + l15;
#pragma unroll
    for (int r = 0; r < 8; ++r) {
        Crow[(size_t)r * N +  0] = acc0[r];
        Crow[(size_t)r * N + 16] = acc1[r];
        Crow[(size_t)r * N + 32] = acc2[r];
        Crow[(size_t)r * N + 48] = acc3[r];
    }
}

// ---------------------------------------------------------------------------
// Edge-weighted scatter-add: agg[dst[e], f] += h[src[e], f] * ew[e]
// One thread per (edge, feature); D = 1<<dlog2. Coalesced within a wave
// (32 lanes = 32 consecutive features of one edge); atomics land in L2.
// ---------------------------------------------------------------------------
__global__ void scatter_edges(const float* __restrict__ h,
                              const int* __restrict__ src,
                              const int* __restrict__ dst,
                              const float* __restrict__ ew,
                              float* __restrict__ agg,
                              unsigned total, int dlog2)
{
    unsigned idx = blockIdx.x * blockDim.x + threadIdx.x;
    if (idx >= total) return;
    unsigned e = idx >> dlog2;
    unsigned f = idx & ((1u << dlog2) - 1u);
    float v = h[((size_t)src[e] << dlog2) + f] * ew[e];
    atomicAdd(&agg[((size_t)dst[e] << dlog2) + f], v);
}

__global__ void zero_f32(float* __restrict__ p, unsigned n)
{
    unsigned i = blockIdx.x * blockDim.x + threadIdx.x;
    if (i < n) p[i] = 0.0f;
}

// layer 0 epilogue: x1 = agg + b ; save x1 (residual anchor); h = x1 + relu(x1)
__global__ void bias_res_act1(const float* __restrict__ agg,
                              const float* __restrict__ b,
                              float* __restrict__ x1_out,
                              float* __restrict__ h_out,
                              unsigned n, unsigned dmask)
{
    unsigned i = blockIdx.x * blockDim.x + threadIdx.x;
    if (i >= n) return;
    float x1 = agg[i] + b[i & dmask];
    x1_out[i] = x1;
    h_out[i]  = x1 + fmaxf(x1, 0.0f);
}

// layer 1 epilogue: h = x1 + relu(agg + b)
__global__ void bias_res_act2(const float* __restrict__ agg,
                              const float* __restrict__ b,
                              const float* __restrict__ x1,
                              float* __restrict__ h_out,
                              unsigned n, unsigned dmask)
{
    unsigned i = blockIdx.x * blockDim.x + threadIdx.x;
    if (i >= n) return;
    float h2 = agg[i] + b[i & dmask];
    h_out[i] = x1[i] + fmaxf(h2, 0.0f);
}

// layer 2 epilogue: out += b (in place)
__global__ void bias_inplace(float* __restrict__ out,
                             const float* __restrict__ b,
                             unsigned n, unsigned dmask)
{
    unsigned i = blockIdx.x * blockDim.x + threadIdx.x;
    if (i >= n) return;
    out[i] += b[i & dmask];
}

// ---------------------------------------------------------------------------
extern "C" void kernel_launch(void* const* d_in, const int* in_sizes, int n_in,
                              void* d_out, int out_size, void* d_ws, size_t ws_size,
                              hipStream_t stream)
{
    const float* x  = (const float*)d_in[0];   // [N,128]
    const int*   ei = (const int*)  d_in[1];   // [2,E]
    const float* ew = (const float*)d_in[2];   // [E]
    const float* W1 = (const float*)d_in[3];   // [128,128]
    const float* b1 = (const float*)d_in[4];   // [128]
    const float* W2 = (const float*)d_in[5];   // [128,128]
    const float* b2 = (const float*)d_in[6];   // [128]
    const float* W3 = (const float*)d_in[7];   // [128,64]
    const float* b3 = (const float*)d_in[8];   // [64]
    float* out = (float*)d_out;                // [N,64]

    const int N = in_sizes[0] / 128;           // 50000 (divisible by 16)
    const int E = in_sizes[2];                 // 800000
    const int* srcI = ei;
    const int* dstI = ei + E;

    // workspace: 3 rotating [N,128] f32 buffers (~77 MB total)
    float* bufA = (float*)d_ws;
    float* bufB = bufA + (size_t)N * 128;
    float* bufC = bufB + (size_t)N * 128;

    const dim3 wave(32);
    const int gemmB128 = (N / 16) * (128 / 64);
    const int gemmB64  = (N / 16) * (64 / 64);

    const unsigned nNode128 = (unsigned)N * 128u;
    const unsigned nNode64  = (unsigned)N * 64u;
    const unsigned nEdge128 = (unsigned)E * 128u;
    const unsigned nEdge64  = (unsigned)E * 64u;
    const unsigned T = 256;
    auto blk = [](unsigned n, unsigned t) { return (n + t - 1) / t; };

    // ----- layer 0: x1 = segsum(X@W1) + b1; h = x1 + relu(x1)
    gemm_wmma_f32<<<gemmB128, wave, 0, stream>>>(x, W1, bufA, N, 128, 128);
    zero_f32<<<blk(nNode128, T), T, 0, stream>>>(bufB, nNode128);
    scatter_edges<<<blk(nEdge128, T), T, 0, stream>>>(bufA, srcI, dstI, ew, bufB, nEdge128, 7);
    bias_res_act1<<<blk(nNode128, T), T, 0, stream>>>(bufB, b1, bufC, bufA, nNode128, 127);

    // ----- layer 1: h = x1 + relu(segsum(h@W2) + b2)
    gemm_wmma_f32<<<gemmB128, wave, 0, stream>>>(bufA, W2, bufB, N, 128, 128);
    zero_f32<<<blk(nNode128, T), T, 0, stream>>>(bufA, nNode128);
    scatter_edges<<<blk(nEdge128, T), T, 0, stream>>>(bufB, srcI, dstI, ew, bufA, nEdge128, 7);
    bias_res_act2<<<blk(nNode128, T), T, 0, stream>>>(bufA, b2, bufC, bufB, nNode128, 127);

    // ----- layer 2: out = segsum(h@W3) + b3
    gemm_wmma_f32<<<gemmB64, wave, 0, stream>>>(bufB, W3, bufA, N, 128, 64);
    zero_f32<<<blk(nNode64, T), T, 0, stream>>>(out, nNode64);
    scatter_edges<<<blk(nEdge64, T), T, 0, stream>>>(bufA, srcI, dstI, ew, out, nEdge64, 6);
    bias_inplace<<<blk(nNode64, T), T, 0, stream>>>(out, b3, nNode64, 63);
}